// MessagePassing_21990232555991
// MI455X (gfx1250) — compile-verified
//
#include <hip/hip_runtime.h>
#include <hip/hip_bf16.h>

typedef __attribute__((ext_vector_type(2))) float v2f;
typedef __attribute__((ext_vector_type(8))) float v8f;

__device__ __forceinline__ float silu_f(float x) {
    return x / (1.0f + __expf(-x));
}

__device__ __forceinline__ v8f wmma_f32_k4(v2f a, v2f b, v8f c) {
    // V_WMMA_F32_16X16X4_F32: D = A(16x4) * B(4x16) + C(16x16)
    return __builtin_amdgcn_wmma_f32_16x16x4_f32(false, a, false, b, (short)0, c,
                                                 false, false);
}

__global__ void k_zero(float* __restrict__ p, int n) {
    int i = blockIdx.x * blockDim.x + threadIdx.x;
    if (i < n) p[i] = 0.0f;
}

// ---------------------------------------------------------------------------
// Node precompute: y0 = x0@Wl0, y1[m] = x1[:,:,m]@Wl1 (lin_norm),
// sc0 = (x0 (x) attrs) @ Wsc0, sc1[m] = (x1m (x) attrs) @ Wsc1 (sc_norm).
// One wave -> one 16x16 output tile via v_wmma_f32_16x16x4_f32.
// 36 problems per node-tile: 4(y0) + 12(y1) + 8(sc0) + 12(sc1)
// ---------------------------------------------------------------------------
__global__ void k_node_pre(const float* __restrict__ nf, const float* __restrict__ attrs,
                           const float* __restrict__ Wl0, const float* __restrict__ Wl1,
                           const float* __restrict__ Wsc0, const float* __restrict__ Wsc1,
                           float* __restrict__ y, float* __restrict__ sc,
                           int N, int ntile) {
    const int wid  = blockIdx.x * (blockDim.x >> 5) + (threadIdx.x >> 5);
    const int tile = wid / 36;
    const int prob = wid % 36;
    if (tile >= ntile) return;            // wave-uniform (EXEC all-1s for WMMA)
    const int lane = threadIdx.x & 31;
    const int r    = lane & 15;
    const int hh   = lane >> 4;
    const int row0 = tile * 16;
    const bool full = (row0 + 16 <= N);   // wave-uniform fast path
    const int ra   = (row0 + r < N) ? (row0 + r) : (N - 1);
    const float lin_norm = 0.125f;                 // 1/sqrt(64)
    const float sc_norm  = 0.04419417382415922f;   // 1/sqrt(512)

    v8f acc = {};
    if (prob < 4) {                                   // ---- y0: x0 @ Wl0
        const int cb = prob * 16;
        const float* arow = nf + (size_t)ra * 256;
        for (int k = 0; k < 64; k += 4) {
            const int ka = k + 2 * hh;
            v2f a = *(const v2f*)(arow + ka);         // contiguous pair -> b64
            v2f b;
            b.x = Wl0[ka * 64 + cb + r];
            b.y = Wl0[(ka + 1) * 64 + cb + r];
            acc = wmma_f32_k4(a, b, acc);
        }
        if (full) {
#pragma unroll
            for (int g = 0; g < 8; ++g)
                y[(row0 + g + 8 * hh) * 256 + cb + r] = acc[g] * lin_norm;
        } else {
            for (int g = 0; g < 8; ++g) {
                const int row = row0 + g + 8 * hh;
                if (row < N) y[row * 256 + cb + r] = acc[g] * lin_norm;
            }
        }
    } else if (prob < 16) {                           // ---- y1[m]: x1m @ Wl1
        const int p  = prob - 4;
        const int m  = p >> 2;
        const int cb = (p & 3) * 16;
        const float* arow = nf + (size_t)ra * 256 + 64 + m;
        for (int k = 0; k < 64; k += 4) {
            const int ka = k + 2 * hh;
            v2f a, b;
            a.x = arow[ka * 3];
            a.y = arow[(ka + 1) * 3];
            b.x = Wl1[ka * 64 + cb + r];
            b.y = Wl1[(ka + 1) * 64 + cb + r];
            acc = wmma_f32_k4(a, b, acc);
        }
        if (full) {
#pragma unroll
            for (int g = 0; g < 8; ++g)
                y[(row0 + g + 8 * hh) * 256 + 64 + (cb + r) * 3 + m] = acc[g] * lin_norm;
        } else {
            for (int g = 0; g < 8; ++g) {
                const int row = row0 + g + 8 * hh;
                if (row < N) y[row * 256 + 64 + (cb + r) * 3 + m] = acc[g] * lin_norm;
            }
        }
    } else if (prob < 24) {                           // ---- sc0: (x0 (x) attrs) @ Wsc0
        const int cb = (prob - 16) * 16;
        // ka = k + 2*hh with k % 8 in {0,4}: only 4 distinct attr values per lane
        const float atA0 = attrs[ra * 8 + 2 * hh];
        const float atA1 = attrs[ra * 8 + 2 * hh + 1];
        const float atB0 = attrs[ra * 8 + 4 + 2 * hh];
        const float atB1 = attrs[ra * 8 + 5 + 2 * hh];
        const float* arow = nf + (size_t)ra * 256;
        for (int k = 0; k < 512; k += 8) {            // u = k>>3; 2 WMMAs per chunk
            const float xu = arow[k >> 3];
            const int ka0 = k + 2 * hh;
            const int ka1 = k + 4 + 2 * hh;
            v2f a, b;
            a.x = xu * atA0;  a.y = xu * atA1;
            b.x = Wsc0[ka0 * 128 + cb + r];
            b.y = Wsc0[(ka0 + 1) * 128 + cb + r];
            acc = wmma_f32_k4(a, b, acc);
            a.x = xu * atB0;  a.y = xu * atB1;
            b.x = Wsc0[ka1 * 128 + cb + r];
            b.y = Wsc0[(ka1 + 1) * 128 + cb + r];
            acc = wmma_f32_k4(a, b, acc);
        }
        if (full) {
#pragma unroll
            for (int g = 0; g < 8; ++g)
                sc[(row0 + g + 8 * hh) * 320 + cb + r] = acc[g] * sc_norm;
        } else {
            for (int g = 0; g < 8; ++g) {
                const int row = row0 + g + 8 * hh;
                if (row < N) sc[row * 320 + cb + r] = acc[g] * sc_norm;
            }
        }
    } else {                                          // ---- sc1[m]: (x1m (x) attrs) @ Wsc1
        const int p  = prob - 24;
        const int m  = p >> 2;
        const int cb = (p & 3) * 16;
        const float atA0 = attrs[ra * 8 + 2 * hh];
        const float atA1 = attrs[ra * 8 + 2 * hh + 1];
        const float atB0 = attrs[ra * 8 + 4 + 2 * hh];
        const float atB1 = attrs[ra * 8 + 5 + 2 * hh];
        const float* arow = nf + (size_t)ra * 256 + 64 + m;
        for (int k = 0; k < 512; k += 8) {
            const float xu = arow[(k >> 3) * 3];
            const int ka0 = k + 2 * hh;
            const int ka1 = k + 4 + 2 * hh;
            v2f a, b;
            a.x = xu * atA0;  a.y = xu * atA1;
            b.x = Wsc1[ka0 * 64 + cb + r];
            b.y = Wsc1[(ka0 + 1) * 64 + cb + r];
            acc = wmma_f32_k4(a, b, acc);
            a.x = xu * atB0;  a.y = xu * atB1;
            b.x = Wsc1[ka1 * 64 + cb + r];
            b.y = Wsc1[(ka1 + 1) * 64 + cb + r];
            acc = wmma_f32_k4(a, b, acc);
        }
        if (full) {
#pragma unroll
            for (int g = 0; g < 8; ++g)
                sc[(row0 + g + 8 * hh) * 320 + 128 + (cb + r) * 3 + m] = acc[g] * sc_norm;
        } else {
            for (int g = 0; g < 8; ++g) {
                const int row = row0 + g + 8 * hh;
                if (row < N) sc[row * 320 + 128 + (cb + r) * 3 + m] = acc[g] * sc_norm;
            }
        }
    }
}

// ---------------------------------------------------------------------------
// Edge kernel: one wave per edge. Only the live messages (m0a,m0b,m1a,m1b)
// are produced: 512 columns per edge, atomically aggregated into agg[dst].
// ---------------------------------------------------------------------------
__global__ void k_edges(const int* __restrict__ eidx, const float* __restrict__ eattr,
                        const float* __restrict__ emb, const float* __restrict__ Wfc1,
                        const float* __restrict__ Wfc2, const float* __restrict__ y,
                        float* __restrict__ agg, int E) {
    const int e = blockIdx.x * (blockDim.x >> 5) + (threadIdx.x >> 5);
    if (e >= E) return;
    const int lane = threadIdx.x & 31;
    const int src  = eidx[e];
    const int dst  = eidx[E + e];
    const float e0 = eattr[e * 4 + 0];
    const float ex = eattr[e * 4 + 1];
    const float ey = eattr[e * 4 + 2];
    const float ez = eattr[e * 4 + 3];
    const float fc_norm = 0.3535533905932738f;   // 1/sqrt(8)
    const float INV3    = 0.5773502691896258f;   // 1/sqrt(3)

    float hb[8];
#pragma unroll
    for (int j = 0; j < 8; ++j) {
        float s = 0.0f;
#pragma unroll
        for (int i = 0; i < 8; ++i) s += emb[e * 8 + i] * Wfc1[i * 8 + j];
        hb[j] = silu_f(s * fc_norm);
    }

    float* aggd = agg + (size_t)dst * 512;
    const float* ys = y + (size_t)src * 256;

#pragma unroll
    for (int t = 0; t < 2; ++t) {
        const int u = lane + 32 * t;
        float w0 = 0.f, w1 = 0.f, w2 = 0.f, w3 = 0.f;
#pragma unroll
        for (int i = 0; i < 8; ++i) {
            const float hv = hb[i];
            const float* row = Wfc2 + i * 384;
            w0 += hv * row[u];
            w1 += hv * row[64 + u];
            w2 += hv * row[128 + u];
            w3 += hv * row[192 + u];
        }
        w0 *= fc_norm; w1 *= fc_norm; w2 *= fc_norm; w3 *= fc_norm;

        const float x0  = ys[u];
        const float x1x = ys[64 + u * 3 + 0];
        const float x1y = ys[64 + u * 3 + 1];
        const float x1z = ys[64 + u * 3 + 2];

        unsafeAtomicAdd(&aggd[u],      w0 * x0 * e0);
        unsafeAtomicAdd(&aggd[64 + u], w3 * (x1x * ex + x1y * ey + x1z * ez) * INV3);
        const float wa = w1 * x0;
        unsafeAtomicAdd(&aggd[128 + u * 3 + 0], wa * ex);
        unsafeAtomicAdd(&aggd[128 + u * 3 + 1], wa * ey);
        unsafeAtomicAdd(&aggd[128 + u * 3 + 2], wa * ez);
        const float wb = w2 * e0;
        unsafeAtomicAdd(&aggd[320 + u * 3 + 0], wb * x1x);
        unsafeAtomicAdd(&aggd[320 + u * 3 + 1], wb * x1y);
        unsafeAtomicAdd(&aggd[320 + u * 3 + 2], wb * x1z);
    }
}

// ---------------------------------------------------------------------------
// lin2: out0 = agg0@W20*norm + sc0 ; out1[m] = agg1m@W21*norm + sc1[m]
// 20 WMMA tile-problems per node tile: 8(out0) + 12(out1)
// ---------------------------------------------------------------------------
__global__ void k_lin2(const float* __restrict__ agg, const float* __restrict__ sc,
                       const float* __restrict__ W20, const float* __restrict__ W21,
                       float* __restrict__ o0, float* __restrict__ o1,
                       int N, int ntile) {
    const int wid  = blockIdx.x * (blockDim.x >> 5) + (threadIdx.x >> 5);
    const int tile = wid / 20;
    const int prob = wid % 20;
    if (tile >= ntile) return;
    const int lane = threadIdx.x & 31;
    const int r    = lane & 15;
    const int hh   = lane >> 4;
    const int row0 = tile * 16;
    const bool full = (row0 + 16 <= N);
    const int ra   = (row0 + r < N) ? (row0 + r) : (N - 1);
    const float lin2_norm = 0.08838834764831845f;  // 1/sqrt(128)

    v8f acc = {};
    if (prob < 8) {                                   // ---- out0
        const int cb = prob * 16;
        const float* arow = agg + (size_t)ra * 512;
        for (int k = 0; k < 128; k += 4) {
            const int ka = k + 2 * hh;
            v2f a = *(const v2f*)(arow + ka);         // contiguous pair -> b64
            v2f b;
            b.x = W20[ka * 128 + cb + r];
            b.y = W20[(ka + 1) * 128 + cb + r];
            acc = wmma_f32_k4(a, b, acc);
        }
        if (full) {
#pragma unroll
            for (int g = 0; g < 8; ++g) {
                const int row = row0 + g + 8 * hh;
                o0[row * 128 + cb + r] = acc[g] * lin2_norm + sc[row * 320 + cb + r];
            }
        } else {
            for (int g = 0; g < 8; ++g) {
                const int row = row0 + g + 8 * hh;
                if (row < N)
                    o0[row * 128 + cb + r] = acc[g] * lin2_norm + sc[row * 320 + cb + r];
            }
        }
    } else {                                          // ---- out1[m]
        const int p  = prob - 8;
        const int m  = p >> 2;
        const int cb = (p & 3) * 16;
        const float* arow = agg + (size_t)ra * 512 + 128 + m;
        for (int k = 0; k < 128; k += 4) {
            const int ka = k + 2 * hh;
            v2f a, b;
            a.x = arow[ka * 3];
            a.y = arow[(ka + 1) * 3];
            b.x = W21[ka * 64 + cb + r];
            b.y = W21[(ka + 1) * 64 + cb + r];
            acc = wmma_f32_k4(a, b, acc);
        }
        if (full) {
#pragma unroll
            for (int g = 0; g < 8; ++g) {
                const int row = row0 + g + 8 * hh;
                const int c   = cb + r;
                o1[row * 192 + c * 3 + m] =
                    acc[g] * lin2_norm + sc[row * 320 + 128 + c * 3 + m];
            }
        } else {
            for (int g = 0; g < 8; ++g) {
                const int row = row0 + g + 8 * hh;
                if (row < N) {
                    const int c = cb + r;
                    o1[row * 192 + c * 3 + m] =
                        acc[g] * lin2_norm + sc[row * 320 + 128 + c * 3 + m];
                }
            }
        }
    }
}

// ---------------------------------------------------------------------------
// Gating + residual
// ---------------------------------------------------------------------------
__global__ void k_gate(const float* __restrict__ o0, const float* __restrict__ o1,
                       const float* __restrict__ nf, float* __restrict__ out, int N) {
    const int n = blockIdx.x;
    const int w = threadIdx.x;
    if (n >= N || w >= 64) return;
    const float s = silu_f(o0[n * 128 + w]);
    const float g = silu_f(o0[n * 128 + 64 + w]);
    out[n * 256 + w] = s + nf[n * 256 + w];
#pragma unroll
    for (int m = 0; m < 3; ++m) {
        const int c = 64 + w * 3 + m;
        out[n * 256 + c] = g * o1[n * 192 + w * 3 + m] + nf[n * 256 + c];
    }
}

extern "C" void kernel_launch(void* const* d_in, const int* in_sizes, int n_in,
                              void* d_out, int out_size, void* d_ws, size_t ws_size,
                              hipStream_t stream) {
    const float* nf    = (const float*)d_in[0];
    const float* attrs = (const float*)d_in[1];
    const int*   eidx  = (const int*)d_in[2];
    const float* eattr = (const float*)d_in[3];
    const float* emb   = (const float*)d_in[4];
    const float* Wl0   = (const float*)d_in[5];
    const float* Wl1   = (const float*)d_in[6];
    const float* Wfc1  = (const float*)d_in[7];
    const float* Wfc2  = (const float*)d_in[8];
    const float* W20   = (const float*)d_in[9];
    const float* W21   = (const float*)d_in[10];
    const float* Wsc0  = (const float*)d_in[11];
    const float* Wsc1  = (const float*)d_in[12];
    float* out = (float*)d_out;

    const int N = in_sizes[0] / 256;
    const int E = in_sizes[2] / 2;

    float* ws  = (float*)d_ws;
    float* y   = ws;                         // N*256
    float* sc  = ws + (size_t)N * 256;       // N*320
    float* agg = ws + (size_t)N * 576;       // N*512
    float* o0  = ws + (size_t)N * 1088;      // N*128
    float* o1  = ws + (size_t)N * 1216;      // N*192

    const int ntile = (N + 15) / 16;

    {   // zero the aggregation buffer (atomics accumulate into it)
        const int total = N * 512;
        k_zero<<<(total + 255) / 256, 256, 0, stream>>>(agg, total);
    }
    {   // node precompute (WMMA f32)
        const int waves  = ntile * 36;
        const int blocks = (waves + 7) / 8;
        k_node_pre<<<blocks, 256, 0, stream>>>(nf, attrs, Wl0, Wl1, Wsc0, Wsc1,
                                               y, sc, N, ntile);
    }
    {   // per-edge messages + scatter-add (L2-resident atomics)
        const int blocks = (E + 7) / 8;
        k_edges<<<blocks, 256, 0, stream>>>(eidx, eattr, emb, Wfc1, Wfc2, y, agg, E);
    }
    {   // lin2 (WMMA f32) + skip connection
        const int waves  = ntile * 20;
        const int blocks = (waves + 7) / 8;
        k_lin2<<<blocks, 256, 0, stream>>>(agg, sc, W20, W21, o0, o1, N, ntile);
    }
    // gating + residual
    k_gate<<<N, 64, 0, stream>>>(o0, o1, nf, out, N);
}